// SpatialTransformer_8675833938223
// MI455X (gfx1250) — compile-verified
//
#include <hip/hip_runtime.h>

// ---------------------------------------------------------------------------
// 3D spatial transformer (trilinear warp, border padding) for MI455X/gfx1250.
//
// Memory-bound target: flow 59MB (async-staged, NT) + src ~20MB (L2-resident
// gathers) + out 20MB (NT store) ~= 100MB => ~4.3us at 23.3 TB/s. The identity
// grid input is reconstructed from voxel indices (it is exactly meshgrid).
//
// This revision attacks VALU cost so vector-ALU time stays under the memory
// roofline: 32-bit indices everywhere, saddr-form async loads (no per-lane
// 64-bit address chains), division-free incremental (w,h,d) tracking in f32,
// and a fully unrolled double-buffered async pipeline.
// ---------------------------------------------------------------------------

namespace {
constexpr int Dd = 160, Hh = 192, Ww = 160;
constexpr int HW = Hh * Ww;              // 30720
constexpr int NVOX = Dd * HW;            // 4,915,200 (fits easily in i32)
constexpr int BLOCK = 256;               // 8 wave32s
constexpr int ITER  = 8;
constexpr int SPAN  = BLOCK * ITER;      // 2048; NVOX = 2400 * SPAN exactly
}

// Per-lane async dword copy global->LDS (gfx1250, ASYNCcnt-tracked), saddr
// form: 64-bit uniform base in SGPRs + 32-bit per-lane byte offset in a VGPR.
// NT hint: flow data is single-use streaming; don't displace src in L2.
__device__ __forceinline__ void async_ld_f32(unsigned lds_off, unsigned voff_bytes,
                                             const float* sbase) {
  asm volatile("global_load_async_to_lds_b32 %0, %1, %2 th:TH_LOAD_NT"
               :
               : "v"(lds_off), "v"(voff_bytes), "s"(sbase)
               : "memory");
}

__global__ __launch_bounds__(BLOCK)
void SpatialTransformer_warp3d(const float* __restrict__ src,
                               const float* __restrict__ flow,
                               float* __restrict__ out) {
  // Double-buffered staging for the 3 flow planes (6 KB LDS).
  __shared__ float fbuf[2][3][BLOCK];

  const int tid   = threadIdx.x;
  const int start = (int)blockIdx.x * SPAN + tid;   // this thread's voxel 0

  const float* fDp = flow;                // displacement along D (z)
  const float* fHp = flow + NVOX;         // displacement along H (y)
  const float* fWp = flow + 2 * NVOX;     // displacement along W (x)

  // u = (index + flow) * S/(S-1) - 0.5 (== reference's normalize/unnormalize).
  constexpr float sclx = (float)Ww / (float)(Ww - 1);
  constexpr float scly = (float)Hh / (float)(Hh - 1);
  constexpr float sclz = (float)Dd / (float)(Dd - 1);

  // One-time decomposition of the starting flat index; thereafter (w,h,d)
  // advance incrementally (+256 voxels = +96 in w, +1 or +2 in h, carry to d).
  // Tracked in f32: all values are small exact integers.
  const unsigned s0 = (unsigned)start;
  float wf = (float)(int)(s0 % (unsigned)Ww);
  const unsigned t0 = s0 / (unsigned)Ww;
  float hf = (float)(int)(t0 % (unsigned)Hh);
  float df = (float)(int)(t0 / (unsigned)Hh);

  auto ldsoff = [&](int b, int p) -> unsigned {
    return (unsigned)(size_t)&fbuf[b][p][tid];   // low 32 bits = LDS byte addr
  };

  // Prologue: stage iteration 0 (ASYNCcnt = 3).
  {
    const unsigned voff = (unsigned)start << 2;
    async_ld_f32(ldsoff(0, 0), voff, fDp);
    async_ld_f32(ldsoff(0, 1), voff, fHp);
    async_ld_f32(ldsoff(0, 2), voff, fWp);
  }

#pragma unroll
  for (int it = 0; it < ITER; ++it) {
    const int buf = it & 1;

    if (it + 1 < ITER) {
      // Stage next chunk, then wait until only those 3 newest ops remain.
      // Per-wave async loads complete in order => current buffer has landed.
      const unsigned voff = (unsigned)(start + (it + 1) * BLOCK) << 2;
      async_ld_f32(ldsoff(buf ^ 1, 0), voff, fDp);
      async_ld_f32(ldsoff(buf ^ 1, 1), voff, fHp);
      async_ld_f32(ldsoff(buf ^ 1, 2), voff, fWp);
      asm volatile("s_wait_asynccnt 3" ::: "memory");
    } else {
      asm volatile("s_wait_asynccnt 0" ::: "memory");
    }

    const float fD = fbuf[buf][0][tid];
    const float fH = fbuf[buf][1][tid];
    const float fW = fbuf[buf][2][tid];

    // Sample coordinates (grid_sample align_corners=False convention).
    float ux = fmaf(wf + fW, sclx, -0.5f);
    float uy = fmaf(hf + fH, scly, -0.5f);
    float uz = fmaf(df + fD, sclz, -0.5f);

    // Border padding: clamp into [0, S-1].
    ux = fminf(fmaxf(ux, 0.0f), (float)(Ww - 1));
    uy = fminf(fmaxf(uy, 0.0f), (float)(Hh - 1));
    uz = fminf(fmaxf(uz, 0.0f), (float)(Dd - 1));

    const float xf = floorf(ux), yf = floorf(uy), zf = floorf(uz);
    const float fx = ux - xf, fy = uy - yf, fz = uz - zf;

    const int x0 = (int)xf, y0 = (int)yf, z0 = (int)zf;
    const int x1 = min(x0 + 1, Ww - 1);
    const int y1 = min(y0 + 1, Hh - 1);
    const int z1 = min(z0 + 1, Dd - 1);

    const int zo0 = z0 * HW, zo1 = z1 * HW;
    const int yo0 = y0 * Ww, yo1 = y1 * Ww;

    // 8-point gather (volume is L2-resident; displacements ~N(0,3) voxels).
    const float c000 = src[zo0 + yo0 + x0];
    const float c001 = src[zo0 + yo0 + x1];
    const float c010 = src[zo0 + yo1 + x0];
    const float c011 = src[zo0 + yo1 + x1];
    const float c100 = src[zo1 + yo0 + x0];
    const float c101 = src[zo1 + yo0 + x1];
    const float c110 = src[zo1 + yo1 + x0];
    const float c111 = src[zo1 + yo1 + x1];

    // Same weight-product form as the reference.
    const float wx1 = fx, wx0 = 1.0f - fx;
    const float wy1 = fy, wy0 = 1.0f - fy;
    const float wz1 = fz, wz0 = 1.0f - fz;
    const float r =
        wz0 * (wy0 * (wx0 * c000 + wx1 * c001) + wy1 * (wx0 * c010 + wx1 * c011)) +
        wz1 * (wy0 * (wx0 * c100 + wx1 * c101) + wy1 * (wx0 * c110 + wx1 * c111));

    // Output written once, never re-read: non-temporal store.
    __builtin_nontemporal_store(r, out + (start + it * BLOCK));

    // Advance (w,h,d) by 256 voxels: 256 = 160 + 96.
    wf += 96.0f;
    hf += 1.0f;
    if (wf >= (float)Ww) { wf -= (float)Ww; hf += 1.0f; }
    if (hf >= (float)Hh) { hf -= (float)Hh; df += 1.0f; }
  }
}

extern "C" void kernel_launch(void* const* d_in, const int* in_sizes, int n_in,
                              void* d_out, int out_size, void* d_ws, size_t ws_size,
                              hipStream_t stream) {
  const float* src  = (const float*)d_in[0];   // [1,1,D,H,W] f32
  const float* flow = (const float*)d_in[1];   // [1,3,D,H,W] f32
  // d_in[2] (identity meshgrid) is reconstructed analytically in-kernel.
  float* out = (float*)d_out;                  // [1,1,D,H,W] f32

  const unsigned nblocks = (unsigned)(NVOX / SPAN);  // 2400, exact
  SpatialTransformer_warp3d<<<dim3(nblocks), dim3(BLOCK), 0, stream>>>(src, flow, out);
}